// OrthogonalMaps_34248069219258
// MI455X (gfx1250) — compile-verified
//
#include <hip/hip_runtime.h>

// CDNA5 / gfx1250. Batched 16x16 Householder QR -> Q, WY form on fp32 WMMA.
//
// Q = I - V T V^T  (LAPACK geqrf/orgqr convention, matches jnp.linalg.qr)
// Computed as  M = V * (-T)  then  Q = M * V^T + I,
// each a 16x16x16 fp32 product = chain of 4x v_wmma_f32_16x16x4_f32.

typedef __attribute__((ext_vector_type(2))) float v2f;
typedef __attribute__((ext_vector_type(8))) float v8f;

#define WAVES_PER_BLOCK 8
#define LDS_PER_WAVE 1536  // floats: V[2*256] + Tn[2*256] + M[256] + Q[256]

__global__ __launch_bounds__(256) void ortho_qr_kernel(
    const float* __restrict__ params, float* __restrict__ out, int nmat) {
  const int lane = threadIdx.x & 31;
  const int wave = threadIdx.x >> 5;
  const int jj   = lane & 15;        // column (QR phase) / N index (WMMA phase)
  const int h    = (lane >> 4) & 1;  // which matrix of the pair (QR phase)
  const int hb   = 2 * h;            // K sub-offset for hi half-wave (WMMA A/B layout)

  const int wid  = blockIdx.x * WAVES_PER_BLOCK + wave;
  const int mat0 = wid * 2;
  int mymat = mat0 + h;
  if (mymat >= nmat) mymat = nmat - 1;  // clamp: loads stay in-bounds, stores are predicated

  __shared__ __align__(16) float lds[WAVES_PER_BLOCK * LDS_PER_WAVE];
  float* Vb = lds + wave * LDS_PER_WAVE;  // V col-major: Vb[mm*256 + col*16 + row]
  float* Tb = Vb + 512;                   // -T row-major: Tb[mm*256 + row*16 + col]
  float* Mb = Tb + 512;                   // M row-major (per-wave scratch, reused per mm)
  float* Qb = Mb + 256;                   // Q row-major staging

  // ---------------- load A column jj: A = I + strict_lower(params) ----------------
  // params index of A[i][j] (i>j) is i*(i-1)/2 + j  (tril_indices(16,-1) order).
  float a[16];
  {
    const size_t pb = (size_t)mymat * 120;
    a[0] = (jj == 0) ? 1.f : 0.f;
#pragma unroll
    for (int i = 1; i < 16; ++i) {
      int idx = i * (i - 1) / 2 + jj;
      if (idx > 119) idx = 119;  // keep inactive lanes in-bounds
      float p = params[pb + idx];
      a[i] = (i > jj) ? p : ((i == jj) ? 1.f : 0.f);
    }
  }

  // ---------------- Householder QR + forward-columnwise T (stored negated) --------
  // Lane j keeps column j; after step j it holds the reflector v_j.
  // Lane i also accumulates row i of Tn = -T.
  float t[16];
#pragma unroll
  for (int k = 0; k < 16; ++k) {
    // broadcast current column k within each 16-lane half
    float col[16];
#pragma unroll
    for (int i = 0; i < 16; ++i)
      if (i >= k) col[i] = __shfl(a[i], k, 16);

    float alpha = col[k];
    float sigma = 0.f;
#pragma unroll
    for (int i = 0; i < 16; ++i)
      if (i > k) sigma += col[i] * col[i];

    const bool zdeg = (sigma == 0.f);                 // LAPACK larfg: tau = 0
    float d    = sqrtf(alpha * alpha + sigma);
    float beta = (alpha >= 0.f) ? -d : d;
    float tau  = zdeg ? 0.f : (beta - alpha) / beta;
    float inv  = zdeg ? 0.f : 1.f / (alpha - beta);

    float v[16];
#pragma unroll
    for (int i = 0; i < 16; ++i)
      v[i] = (i < k) ? 0.f : ((i == k) ? 1.f : col[i] * inv);

    // z_j = v_k . a_j over rows k..15 (exact: v_k is zero above row k).
    // For j<k this is v_j^T v_k (T recurrence); for j>k it is the update projection.
    float z = 0.f;
#pragma unroll
    for (int i = 0; i < 16; ++i)
      if (i >= k) z += a[i] * v[i];

    // Tn[jj][k]:  -T[0:k,k] = tau * T[0:k,0:k] * z[0:k]  =>  -tau * sum_j Tn[jj][j]*z_j
    float acc = 0.f;
#pragma unroll
    for (int j = 0; j < 16; ++j)
      if (j < k) { float zb = __shfl(z, j, 16); acc += t[j] * zb; }
    t[k] = (jj < k) ? (-tau * acc) : ((jj == k) ? -tau : 0.f);

    // trailing-column update; lane k swaps its column for v_k
#pragma unroll
    for (int i = 0; i < 16; ++i)
      if (i >= k) {
        float upd = a[i] - tau * z * v[i];
        a[i] = (jj > k) ? upd : ((jj == k) ? v[i] : a[i]);
      }
  }

  // publish V (zeroed above diagonal) and Tn
#pragma unroll
  for (int i = 0; i < 16; ++i) {
    Vb[h * 256 + jj * 16 + i] = (i < jj) ? 0.f : a[i];  // col-major V
    Tb[h * 256 + jj * 16 + i] = t[i];                   // row-major -T
  }
  __syncthreads();

  // ---------------- Q = (V * -T) * V^T + I  via fp32 WMMA chains ------------------
  // A 16x4 layout: VGPR0 = {K0 | K2}, VGPR1 = {K1 | K3}; lanes 0-15 M=lane, 16-31 M=lane-16.
  // B 4x16 layout mirrors A: VGPR0 = {row K0 | row K2}, VGPR1 = {row K1 | row K3}.
  // C/D: VGPR r -> M = r + 8*(lane>=16), N = lane&15.
#pragma unroll
  for (int mm = 0; mm < 2; ++mm) {
    const float* Vm = Vb + mm * 256;
    const float* Tm = Tb + mm * 256;

    v8f m = {0.f, 0.f, 0.f, 0.f, 0.f, 0.f, 0.f, 0.f};
#pragma unroll
    for (int kk = 0; kk < 4; ++kk) {
      const int kb = kk * 4 + hb;
      v2f af, bf;
      af.x = Vm[(kb + 0) * 16 + jj];  // V[row=jj][kb+0]
      af.y = Vm[(kb + 1) * 16 + jj];
      bf.x = Tm[(kb + 0) * 16 + jj];  // Tn[kb+0][col=jj]
      bf.y = Tm[(kb + 1) * 16 + jj];
      m = __builtin_amdgcn_wmma_f32_16x16x4_f32(false, af, false, bf,
                                                (short)0, m, false, false);
    }
#pragma unroll
    for (int r = 0; r < 8; ++r)
      Mb[(r + 8 * h) * 16 + jj] = m[r];  // spill M row-major for A-operand re-layout
    __syncthreads();

    v8f q;
#pragma unroll
    for (int r = 0; r < 8; ++r)
      q[r] = ((r + 8 * h) == jj) ? 1.f : 0.f;  // C = I
#pragma unroll
    for (int kk = 0; kk < 4; ++kk) {
      const int kb = kk * 4 + hb;
      v2f af, bf;
      af.x = Mb[jj * 16 + kb + 0];     // M[row=jj][kb+0]
      af.y = Mb[jj * 16 + kb + 1];
      bf.x = Vm[(kb + 0) * 16 + jj];   // (V^T)[kb+0][col=jj] = V[jj][kb+0]
      bf.y = Vm[(kb + 1) * 16 + jj];
      q = __builtin_amdgcn_wmma_f32_16x16x4_f32(false, af, false, bf,
                                                (short)0, q, false, false);
    }
#pragma unroll
    for (int r = 0; r < 8; ++r)
      Qb[(r + 8 * h) * 16 + jj] = q[r];
    __syncthreads();

    // coalesced 1KB store of Q (wave-uniform predicate)
    const int mat = mat0 + mm;
    if (mat < nmat) {
      const float4* Qv = reinterpret_cast<const float4*>(Qb);
      float4* Ov = reinterpret_cast<float4*>(out + (size_t)mat * 256);
      Ov[lane * 2 + 0] = Qv[lane * 2 + 0];
      Ov[lane * 2 + 1] = Qv[lane * 2 + 1];
    }
    __syncthreads();
  }
}

extern "C" void kernel_launch(void* const* d_in, const int* in_sizes, int n_in,
                              void* d_out, int out_size, void* d_ws, size_t ws_size,
                              hipStream_t stream) {
  (void)n_in; (void)out_size; (void)d_ws; (void)ws_size;
  const float* params = (const float*)d_in[0];
  float* out = (float*)d_out;
  const int nmat = in_sizes[0] / 120;             // 500000 for the reference shapes
  const int waves = (nmat + 1) / 2;               // 2 matrices per wave
  const int blocks = (waves + WAVES_PER_BLOCK - 1) / WAVES_PER_BLOCK;
  if (blocks > 0) {
    ortho_qr_kernel<<<blocks, 32 * WAVES_PER_BLOCK, 0, stream>>>(params, out, nmat);
  }
}